// GCN_49271864820055
// MI455X (gfx1250) — compile-verified
//
#include <hip/hip_runtime.h>
#include <math.h>

typedef __attribute__((ext_vector_type(2))) float v2f;
typedef __attribute__((ext_vector_type(8))) float v8f;

#define N_NODES 203          // 7 * 29
#define NPC 29               // nodes per client
#define M_PAD 208            // 13 * 16
#define K_PAD 32             // 8 * 4
#define BATCH 2048
#define ODIM 256
#define IN_ROW_STRIDE (BATCH * ODIM)      // 524288 floats per input row
#define OUT_B_STRIDE (N_NODES * ODIM)     // 51968 floats per batch slab

// ---------------------------------------------------------------------------
// Kernel 1: build the zero-padded 208x32 A matrix (the Laplacian column slice)
//   col[n][k] = d[n] * (adj[n][id*29+k] + (n==id*29+k)) * d[id*29+k]
//   d[n] = 1/sqrt(1 + sum_m adj[n][m])
// One block, 256 threads. Tiny (203x203 read once).
// ---------------------------------------------------------------------------
__global__ __launch_bounds__(256) void GCN_prep_kernel(const float* __restrict__ adj,
                                                       const int* __restrict__ idp,
                                                       float* __restrict__ colPad) {
    __shared__ float sd[N_NODES];
    const int t = threadIdx.x;
    if (t < N_NODES) {
        float sum = 1.0f;                         // self loop
        for (int m = 0; m < N_NODES; ++m) sum += adj[t * N_NODES + m];
        sd[t] = 1.0f / sqrtf(sum);
    }
    __syncthreads();
    const int base = idp[0] * NPC;                // 3*29 = 87
    for (int idx = t; idx < M_PAD * K_PAD; idx += 256) {
        const int n = idx >> 5;
        const int k = idx & 31;
        float v = 0.0f;
        if (n < N_NODES && k < NPC) {
            const int m = base + k;
            float a = adj[n * N_NODES + m] + (n == m ? 1.0f : 0.0f);
            v = sd[n] * a * sd[m];
        }
        colPad[idx] = v;
    }
}

// ---------------------------------------------------------------------------
// Kernel 2: per-batch GEMM via V_WMMA_F32_16X16X4_F32.
// Block = one batch b. 8 waves; wave w owns N-tiles {2w, 2w+1}.
// A (208x32, zero-padded) in LDS; per-lane layout per ISA 7.12.2:
//   A: lane&15 = M row, lane>=16 selects K-offset {2,3} vs {0,1}  -> ds_load_b64
//   B: lane&15 = N col, lane>=16 selects K rows {2,3} vs {0,1}
//   C/D: VGPR r holds M = r + (lane>=16 ? 8 : 0), N = lane&15
// ---------------------------------------------------------------------------
__global__ __launch_bounds__(256) void GCN_gemm_kernel(const float* __restrict__ colPad,
                                                       const float* __restrict__ inputs,
                                                       float* __restrict__ out) {
    __shared__ float sA[M_PAD * K_PAD];           // 26624 B

    // Stage A into LDS (26 dwords per thread).
    for (int idx = threadIdx.x; idx < M_PAD * K_PAD; idx += 256) sA[idx] = colPad[idx];
    __syncthreads();

    const int w    = threadIdx.x >> 5;            // wave id 0..7 (wave32)
    const int lane = threadIdx.x & 31;
    const int half = lane >> 4;                   // 0 or 1
    const int n0   = lane & 15;
    const int koff = half << 1;                   // 0 or 2
    const unsigned b = blockIdx.x;

    for (int t = 0; t < 2; ++t) {
        const int j = w * 2 + t;                  // N-tile 0..15

        // ---- Load B tile: 8 steps x {K=koff, K=koff+1}, reused by 13 M-tiles.
        // Clamp K to 28 (valid row): padded A columns are zero so the value
        // is irrelevant but stays finite (no 0*Inf NaN) and in-bounds.
        v2f B8[8];
        const unsigned cbase = b * (unsigned)ODIM + (unsigned)(j * 16 + n0);
#pragma unroll
        for (int s = 0; s < 8; ++s) {
            int k0 = 4 * s + koff;
            int k1 = k0 + 1;
            k0 = k0 > 28 ? 28 : k0;
            k1 = k1 > 28 ? 28 : k1;
            v2f bb;
            bb.x = inputs[(size_t)k0 * IN_ROW_STRIDE + cbase];
            bb.y = inputs[(size_t)k1 * IN_ROW_STRIDE + cbase];
            B8[s] = bb;
        }

        // ---- 13 M-tiles, each: 8 chained fp32 WMMAs over K=32 (29 real).
        for (int i = 0; i < 13; ++i) {
            v8f acc = {};
            const float* rowp = &sA[(i * 16 + n0) * K_PAD + koff];
#pragma unroll
            for (int s = 0; s < 8; ++s) {
                v2f a = *(const v2f*)(rowp + 4 * s);   // contiguous K pair, 8B aligned
                acc = __builtin_amdgcn_wmma_f32_16x16x4_f32(
                          /*neg_a=*/false, a, /*neg_b=*/false, B8[s],
                          /*c_mod=*/(short)0, acc,
                          /*reuse_a=*/false, /*reuse_b=*/false);
            }
            // ---- Store 16x16 tile; predicate the M=203 edge.
            const int mbase = i * 16 + half * 8;
            const size_t obase = (size_t)b * OUT_B_STRIDE + (size_t)(j * 16 + n0);
#pragma unroll
            for (int r = 0; r < 8; ++r) {
                const int mg = mbase + r;
                if (mg < N_NODES) out[obase + (size_t)mg * ODIM] = acc[r];
            }
        }
    }
}

extern "C" void kernel_launch(void* const* d_in, const int* in_sizes, int n_in,
                              void* d_out, int out_size, void* d_ws, size_t ws_size,
                              hipStream_t stream) {
    const float* adj    = (const float*)d_in[0];   // 203*203 fp32
    const float* inputs = (const float*)d_in[1];   // 29*524288 fp32
    const int*   idp    = (const int*)d_in[4];     // scalar id
    float* out = (float*)d_out;                    // 2048*203*256 fp32
    float* colPad = (float*)d_ws;                  // 208*32 fp32 scratch

    GCN_prep_kernel<<<1, 256, 0, stream>>>(adj, idp, colPad);
    GCN_gemm_kernel<<<BATCH, 256, 0, stream>>>(colPad, inputs, out);
}